// PaDiM_84696755077505
// MI455X (gfx1250) — compile-verified
//
#include <hip/hip_runtime.h>
#include <hip/hip_bf16.h>

// Problem constants (from reference setup)
#define Bn    64          // batch
#define CORIG 448         // original channels
#define Cn    100         // selected channels
#define CP    112         // padded channels (multiple of 16)
#define DSTR  113         // D row stride in LDS (gcd(113,64)=1 -> bank-conflict free)
#define Hn    56
#define Pn    (Hn * Hn)   // 3136 pixels
#define OUTSZ 224

typedef float v2f __attribute__((ext_vector_type(2)));
typedef float v8f __attribute__((ext_vector_type(8)));

// ---------------------------------------------------------------------------
// Fused kernel: per-pixel gather + mean-subtract + quadratic form via f32 WMMA.
// One workgroup (8 wave32) per pixel p:
//   Dld = feats[:, sel, p] - mean[p]  (64 x 112, zero-padded)  -> LDS
//   Sld = icov[p] zero-padded to 112x112                        -> LDS
//   M = D @ S tiled as 4x7 16x16 C-tiles, K in steps of 4
//       (v_wmma_f32_16x16x4_f32), all fragments fed from LDS,
//   dist[b] = sum_j M[b,j] * D[b,j] via shfl reduction + LDS atomics.
// icov is read from HBM exactly once (batched over all 64 rows per pixel).
// ---------------------------------------------------------------------------
__global__ __launch_bounds__(256) void padim_dist_kernel(
    const float* __restrict__ feats, const float* __restrict__ mean,
    const int* __restrict__ sel, const float* __restrict__ icov,
    float* __restrict__ distws) {
  __shared__ float Sld[CP * CP];     // 50176 B
  __shared__ float Dld[Bn * DSTR];   // 28928 B
  __shared__ float distLd[Bn];       //   256 B

  const int p = blockIdx.x;
  const int t = threadIdx.x;

  // Stage D: fused channel-gather + mean subtract, zero-padded cols 100..111.
  for (int idx = t; idx < Bn * CP; idx += 256) {
    int b = idx / CP, c = idx % CP;
    float v = 0.0f;
    if (c < Cn) {
      v = feats[(size_t)b * CORIG * Pn + (size_t)sel[c] * Pn + p] -
          mean[(size_t)p * Cn + c];
    }
    Dld[b * DSTR + c] = v;
  }

  // Stage S: padded inverse-covariance tile (read once per pixel from HBM).
  const float* icovp = icov + (size_t)p * Cn * Cn;
  for (int idx = t; idx < CP * CP; idx += 256) {
    int r = idx / CP, c = idx % CP;
    Sld[idx] = (r < Cn && c < Cn) ? icovp[r * Cn + c] : 0.0f;
  }
  if (t < Bn) distLd[t] = 0.0f;
  __syncthreads();

  const int wave  = t >> 5;
  const int lane  = t & 31;
  const int lhalf = lane & 15;
  const int hi    = lane >> 4;     // 0: lanes 0-15, 1: lanes 16-31

  // 28 output tiles (4 M-tiles x 7 N-tiles) distributed over 8 waves.
  for (int tile = wave; tile < 28; tile += 8) {
    const int m0 = (tile & 3) * 16;   // batch-row tile
    const int n0 = (tile >> 2) * 16;  // channel-col tile
    v8f acc = {};

    // A-fragment row for this lane (ISA layout: lanes 0-15 hold M=0..15).
    const int abase = (m0 + lhalf) * DSTR;

    for (int kk = 0; kk < CP; kk += 4) {
      // A 16x4 f32: VGPR0 = K=0 | K=2 ; VGPR1 = K=1 | K=3
      v2f a, b;
      a.x = Dld[abase + kk + 2 * hi];
      a.y = Dld[abase + kk + 2 * hi + 1];
      // B 4x16 f32: rows striped across lanes, same half-split as A.
      b.x = Sld[(kk + 2 * hi) * CP + n0 + lhalf];
      b.y = Sld[(kk + 2 * hi + 1) * CP + n0 + lhalf];
      acc = __builtin_amdgcn_wmma_f32_16x16x4_f32(
          false, a, false, b, (short)0, acc, false, false);
    }

    // Epilogue: dist[b] += sum_col M[b,col] * D[b,col] for this tile.
    // C/D layout: VGPR i -> rows m0+i (lanes 0-15) / m0+8+i (lanes 16-31).
    const int col = n0 + lhalf;
#pragma unroll
    for (int i = 0; i < 8; ++i) {
      const int row = m0 + i + 8 * hi;
      float v = acc[i] * Dld[row * DSTR + col];
#pragma unroll
      for (int m = 1; m < 16; m <<= 1) v += __shfl_xor(v, m, 32);
      if (lhalf == 0) atomicAdd(&distLd[row], v);
    }
  }
  __syncthreads();

  if (t < Bn) distws[(size_t)t * Pn + p] = distLd[t];
}

// ---------------------------------------------------------------------------
// Bilinear resize (half-pixel centers, edge clamp): (B,56,56) -> (B,224,224).
// ---------------------------------------------------------------------------
__global__ __launch_bounds__(256) void padim_resize_kernel(
    const float* __restrict__ distws, float* __restrict__ out) {
  int idx = blockIdx.x * blockDim.x + threadIdx.x;
  const int total = Bn * OUTSZ * OUTSZ;
  if (idx >= total) return;
  int x = idx % OUTSZ;
  int y = (idx / OUTSZ) % OUTSZ;
  int b = idx / (OUTSZ * OUTSZ);

  const float scale = (float)Hn / (float)OUTSZ;  // 0.25
  float fy = (y + 0.5f) * scale - 0.5f;
  float fx = (x + 0.5f) * scale - 0.5f;
  int y0 = (int)floorf(fy), x0 = (int)floorf(fx);
  float wy = fy - (float)y0, wx = fx - (float)x0;
  int y0c = min(max(y0, 0), Hn - 1), y1c = min(max(y0 + 1, 0), Hn - 1);
  int x0c = min(max(x0, 0), Hn - 1), x1c = min(max(x0 + 1, 0), Hn - 1);

  const float* s = distws + (size_t)b * Pn;
  float v00 = s[y0c * Hn + x0c], v01 = s[y0c * Hn + x1c];
  float v10 = s[y1c * Hn + x0c], v11 = s[y1c * Hn + x1c];
  float v0 = v00 + (v01 - v00) * wx;
  float v1 = v10 + (v11 - v10) * wx;
  out[idx] = v0 + (v1 - v0) * wy;
}

// ---------------------------------------------------------------------------
// Host-side launch. Inputs: feats, mean, icov, sel_idx, image_size.
// Workspace: dist [B][P] f32 (~0.8 MB).
// ---------------------------------------------------------------------------
extern "C" void kernel_launch(void* const* d_in, const int* in_sizes, int n_in,
                              void* d_out, int out_size, void* d_ws,
                              size_t ws_size, hipStream_t stream) {
  const float* feats = (const float*)d_in[0];
  const float* mean  = (const float*)d_in[1];
  const float* icov  = (const float*)d_in[2];
  const int*   sel   = (const int*)d_in[3];
  float*       out   = (float*)d_out;

  float* distws = (float*)d_ws;

  padim_dist_kernel<<<Pn, 256, 0, stream>>>(feats, mean, sel, icov, distws);

  const int total3 = Bn * OUTSZ * OUTSZ;
  padim_resize_kernel<<<(total3 + 255) / 256, 256, 0, stream>>>(distws, out);
}